// AttentionModule_56805237457338
// MI455X (gfx1250) — compile-verified
//
#include <hip/hip_runtime.h>

// ---------------------------------------------------------------------------
// Attention pooling: scores = E@q (N x 64 @ 64), softmax over N, out = w@E.
// Single pass (512MB read once, ~22us floor at 23.3 TB/s), online softmax.
// Weighted-sum GEMV mapped onto V_WMMA_F32_16X16X4_F32; embeddings streamed
// with GLOBAL_LOAD_ASYNC_TO_LDS_B128 through a 4-deep per-wave LDS ring.
// ---------------------------------------------------------------------------

typedef __attribute__((ext_vector_type(2))) float v2f;
typedef __attribute__((ext_vector_type(8))) float v8f;
typedef __attribute__((ext_vector_type(4))) int   v4i;

#define NROWS (1 << 21)     // N = 2097152
#define DCOLS 64
#define NBLK  1024
#define BLOCK 256
#define WAVES (BLOCK / 32)                        // 8
#define ROWS_PER_BLOCK (NROWS / NBLK)             // 2048
#define ROWS_PER_WAVE  (ROWS_PER_BLOCK / WAVES)   // 256
#define NITER (ROWS_PER_WAVE / 4)                 // 64 iterations of 4 rows
#define DEPTH 4                                   // async ring stages

#if __has_builtin(__builtin_amdgcn_global_load_async_to_lds_b128)
#define USE_ASYNC 1
#else
#define USE_ASYNC 0
#endif

#if USE_ASYNC
// Param 1: v4i* in global (addrspace(1), printed as __device__ by clang).
// Param 2: v4i* in LDS (addrspace(3)).
typedef __attribute__((address_space(1))) v4i* as1_v4i;
typedef __attribute__((address_space(3))) v4i* as3_v4i;

// Copy one 1KB stage (4 rows x 64 f32): 2 async B128 ops, 512B coalesced each.
__device__ __forceinline__ void async_copy_stage(const float* g, float* l, int lane) {
  as1_v4i g0 = (as1_v4i)(g + lane * 4);   // 16B per lane
  as3_v4i l0 = (as3_v4i)(l + lane * 4);
  __builtin_amdgcn_global_load_async_to_lds_b128(g0,      l0,      0, 0);
  __builtin_amdgcn_global_load_async_to_lds_b128(g0 + 32, l0 + 32, 0, 0);  // +512B
}
__device__ __forceinline__ void wait_async_le6() { asm volatile("s_wait_asynccnt 0x6" ::: "memory"); }
__device__ __forceinline__ void wait_async_le0() { asm volatile("s_wait_asynccnt 0x0" ::: "memory"); }
__device__ __forceinline__ void wait_ds0()       { asm volatile("s_wait_dscnt 0x0"    ::: "memory"); }
#endif

__global__ __launch_bounds__(BLOCK) void attn_pass1(
    const float* __restrict__ E, const float* __restrict__ q,
    float* __restrict__ wsM, float* __restrict__ wsS, float* __restrict__ wsAcc) {
  const int lane = threadIdx.x & 31;
  const int wave = threadIdx.x >> 5;
  const int nsub = lane & 15;
  const bool lo  = lane < 16;

  const long rowBase = (long)blockIdx.x * ROWS_PER_BLOCK + (long)wave * ROWS_PER_WAVE;

  // Per-lane query values for columns 16*j + nsub (j = 0..3)
  const float qv0 = q[ 0 + nsub];
  const float qv1 = q[16 + nsub];
  const float qv2 = q[32 + nsub];
  const float qv3 = q[48 + nsub];

  // Online-softmax state. acc{j} covers columns 16j..16j+15; only D-row M=0
  // (element [0], lanes 0..15) is live, the other 7 rows stay exactly zero.
  v8f acc0 = {}, acc1 = {}, acc2 = {}, acc3 = {};
  float m = -1e30f;
  float s = 0.0f;

  // Process 4 rows starting at `base4` (global or LDS stage base).
  auto process4 = [&](const float* base4) {
    // B-matrix (4x16 f32) lane layout: VGPR0 -> K=0 (lanes 0-15) / K=2 (16-31),
    // VGPR1 -> K=1 / K=3; N = lane&15. Half-wave reads rows {0,1} or {2,3}.
    const float* pa = base4 + (lane >> 4) * (2 * DCOLS) + nsub;  // row 0 or 2
    const float* pb = pa + DCOLS;                                // row 1 or 3
    const float b00 = pa[0],  b01 = pa[16], b02 = pa[32], b03 = pa[48];
    const float b10 = pb[0],  b11 = pb[16], b12 = pb[32], b13 = pb[48];

    // --- scores: per-lane partial dots, butterfly within 16-lane halves ---
    float p0 = b00 * qv0 + b01 * qv1 + b02 * qv2 + b03 * qv3;
    float p1 = b10 * qv0 + b11 * qv1 + b12 * qv2 + b13 * qv3;
    p0 += __shfl_xor(p0, 8, 32);  p1 += __shfl_xor(p1, 8, 32);
    p0 += __shfl_xor(p0, 4, 32);  p1 += __shfl_xor(p1, 4, 32);
    p0 += __shfl_xor(p0, 2, 32);  p1 += __shfl_xor(p1, 2, 32);
    p0 += __shfl_xor(p0, 1, 32);  p1 += __shfl_xor(p1, 1, 32);
    const float o0 = __shfl_xor(p0, 16, 32);
    const float o1 = __shfl_xor(p1, 16, 32);
    const float sc0 = lo ? p0 : o0;   // score of row +0 (same in all lanes)
    const float sc1 = lo ? p1 : o1;   // row +1
    const float sc2 = lo ? o0 : p0;   // row +2
    const float sc3 = lo ? o1 : p1;   // row +3

    // --- online softmax update ---
    const float mn = fmaxf(fmaxf(sc0, sc1), fmaxf(sc2, sc3));
    const float m_new = fmaxf(m, mn);
    const float scale = __expf(m - m_new);
    const float w0 = __expf(sc0 - m_new);
    const float w1 = __expf(sc1 - m_new);
    const float w2 = __expf(sc2 - m_new);
    const float w3 = __expf(sc3 - m_new);
    s = s * scale + (w0 + w1 + w2 + w3);
    m = m_new;
    // Rescale only the live element (M=0 row); all other D rows remain zero.
    acc0[0] *= scale; acc1[0] *= scale; acc2[0] *= scale; acc3[0] *= scale;

    // --- A-matrix (16x4 f32): row 0 = (w0,w1,w2,w3), rows 1..15 = 0 ---
    // VGPR0: lane0=A[0][0]=w0, lane16=A[0][2]=w2; VGPR1: lane0=w1, lane16=w3.
    const float wA = lo ? w0 : w2;
    const float wB = lo ? w1 : w3;
    v2f A;
    A[0] = (nsub == 0) ? wA : 0.0f;
    A[1] = (nsub == 0) ? wB : 0.0f;
    v2f B0, B1, B2, B3;
    B0[0] = b00; B0[1] = b10;
    B1[0] = b01; B1[1] = b11;
    B2[0] = b02; B2[1] = b12;
    B3[0] = b03; B3[1] = b13;

    // D = A x B + C : row M=0 accumulates sum_k w_k * E[row+k][16j + n].
    acc0 = __builtin_amdgcn_wmma_f32_16x16x4_f32(false, A, false, B0, (short)0, acc0, false, false);
    acc1 = __builtin_amdgcn_wmma_f32_16x16x4_f32(false, A, false, B1, (short)0, acc1, false, false);
    acc2 = __builtin_amdgcn_wmma_f32_16x16x4_f32(false, A, false, B2, (short)0, acc2, false, false);
    acc3 = __builtin_amdgcn_wmma_f32_16x16x4_f32(false, A, false, B3, (short)0, acc3, false, false);
  };

#if USE_ASYNC
  // 4-deep per-wave ring of 1KB stages in LDS, filled by async DMA loads.
  __shared__ __align__(16) float lbuf[WAVES][DEPTH][4 * DCOLS];  // 32 KB
#pragma unroll
  for (int j = 0; j < DEPTH; ++j)
    async_copy_stage(E + (rowBase + 4L * j) * DCOLS, &lbuf[wave][j][0], lane);

  for (int i = 0; i < NITER; ++i) {
    const bool more = (i <= NITER - 1 - DEPTH);   // stage i+DEPTH exists
    if (more) wait_async_le6();   // stage i landed; 3 stages still in flight
    else      wait_async_le0();   // tail: drain
    process4(&lbuf[wave][i & (DEPTH - 1)][0]);
    if (more) {
      wait_ds0();                 // our ds reads of this slot are done
      async_copy_stage(E + (rowBase + 4L * (i + DEPTH)) * DCOLS,
                       &lbuf[wave][i & (DEPTH - 1)][0], lane);
    }
  }
#else
  for (int i = 0; i < NITER; ++i)
    process4(E + (rowBase + 4L * i) * DCOLS);
#endif

  // --- merge the 8 wave partials within the block via LDS ---
  __shared__ float lm[WAVES];
  __shared__ float ls[WAVES];
  __shared__ float lacc[WAVES][DCOLS];
  if (lane == 0) { lm[wave] = m; ls[wave] = s; }
  if (lo) {
    lacc[wave][ 0 + nsub] = acc0[0];
    lacc[wave][16 + nsub] = acc1[0];
    lacc[wave][32 + nsub] = acc2[0];
    lacc[wave][48 + nsub] = acc3[0];
  }
  __syncthreads();

  const int t = threadIdx.x;
  if (t < DCOLS) {
    float mb = lm[0];
#pragma unroll
    for (int w = 1; w < WAVES; ++w) mb = fmaxf(mb, lm[w]);
    float a = 0.0f, sb = 0.0f;
#pragma unroll
    for (int w = 0; w < WAVES; ++w) {
      const float f = __expf(lm[w] - mb);
      a  += lacc[w][t] * f;
      sb += ls[w] * f;
    }
    wsAcc[(long)blockIdx.x * DCOLS + t] = a;
    if (t == 0) { wsM[blockIdx.x] = mb; wsS[blockIdx.x] = sb; }
  }
}

__global__ __launch_bounds__(256) void attn_pass2(
    const float* __restrict__ wsM, const float* __restrict__ wsS,
    const float* __restrict__ wsAcc, float* __restrict__ out) {
  __shared__ float red[256];
  const int t = threadIdx.x;

  // global max over block partials
  float m = -1e30f;
  for (int b = t; b < NBLK; b += 256) m = fmaxf(m, wsM[b]);
  red[t] = m; __syncthreads();
  for (int k = 128; k > 0; k >>= 1) {
    if (t < k) red[t] = fmaxf(red[t], red[t + k]);
    __syncthreads();
  }
  const float M = red[0];
  __syncthreads();

  // global denominator
  float sp = 0.0f;
  for (int b = t; b < NBLK; b += 256) sp += wsS[b] * __expf(wsM[b] - M);
  red[t] = sp; __syncthreads();
  for (int k = 128; k > 0; k >>= 1) {
    if (t < k) red[t] += red[t + k];
    __syncthreads();
  }
  const float S = red[0];
  __syncthreads();

  // weighted accumulator merge: 4 thread-groups x 64 columns (coalesced)
  const int col = t & 63;
  const int grp = t >> 6;
  float a = 0.0f;
  for (int b = grp; b < NBLK; b += 4) a += wsAcc[b * DCOLS + col] * __expf(wsM[b] - M);
  red[t] = a; __syncthreads();
  if (t < DCOLS) out[t] = (red[t] + red[t + 64] + red[t + 128] + red[t + 192]) / S;
}

extern "C" void kernel_launch(void* const* d_in, const int* in_sizes, int n_in,
                              void* d_out, int out_size, void* d_ws, size_t ws_size,
                              hipStream_t stream) {
  (void)in_sizes; (void)n_in; (void)out_size; (void)ws_size;
  const float* E = (const float*)d_in[0];
  const float* q = (const float*)d_in[1];
  float* out = (float*)d_out;

  float* wsM   = (float*)d_ws;          // NBLK
  float* wsS   = wsM + NBLK;            // NBLK
  float* wsAcc = wsS + NBLK;            // NBLK * 64   (~270 KB total)

  attn_pass1<<<NBLK, BLOCK, 0, stream>>>(E, q, wsM, wsS, wsAcc);
  attn_pass2<<<1, 256, 0, stream>>>(wsM, wsS, wsAcc, out);
}